// NeighborhoodPool_10926396801886
// MI455X (gfx1250) — compile-verified
//
#include <hip/hip_runtime.h>
#include <stdint.h>
#include <math.h>

typedef float v2f __attribute__((ext_vector_type(2)));
typedef float v8f __attribute__((ext_vector_type(8)));

#define OMEGA_F 100.0f

// V_WMMA_F32_16X16X4_F32: D(16x16,f32) = A(16x4,f32) x B(4x16,f32) + C
// A: lane<16 holds row M=lane, VGPR0=K0,VGPR1=K1; lane>=16 holds K2,K3 (ISA 7.12.2)
// B: lane<16 holds col N=lane, K0/K1; lane>=16 holds K2/K3 (transpose-symmetric)
// D: VGPR r: lane<16 -> (M=r, N=lane); lane>=16 -> (M=r+8, N=lane-16)
__device__ __forceinline__ v8f wmma4(v2f a, v2f b, v8f c) {
  return __builtin_amdgcn_wmma_f32_16x16x4_f32(false, a, false, b, (short)0, c,
                                               false, false);
}

__device__ __forceinline__ bool slot_edge(const int* ei, const unsigned char* keep,
                                          int E, int slot, int& s, int& d) {
  if (slot < E) { s = ei[slot]; d = ei[E + slot]; return keep[slot] != 0; }
  s = d = slot - E;           // appended self-loop
  return true;
}

__device__ __forceinline__ void edge_feats(const float* pos, int s, int d,
                                           float& f0, float& f1, float& f2) {
  float r0 = pos[d * 3 + 0] - pos[s * 3 + 0];
  float r1 = pos[d * 3 + 1] - pos[s * 3 + 1];
  float r2 = pos[d * 3 + 2] - pos[s * 3 + 2];
  float ss = r0 * r0 + r1 * r1 + r2 * r2;
  bool good = ss > 0.0f;
  float rho = good ? sqrtf(ss) : 0.0f;
  float theta = good ? atan2f(r1, r0) : 0.0f;
  float cl = r2 / (good ? rho : 1.0f);
  cl = fminf(1.0f, fmaxf(-1.0f, cl));
  float phi = good ? asinf(cl) : 0.0f;
  const float INVPI = 0.3183098861837907f;
  f0 = rho; f1 = theta * INVPI; f2 = phi * INVPI;
}

// ---------------- init / dedup ----------------

__global__ void k_score_init(float* score, const float* g2bias, int N) {
  int i = blockIdx.x * blockDim.x + threadIdx.x;
  if (i < N) score[i] = g2bias[0];
}

__global__ void k_dedup(const int* ei, unsigned int* bitmap, unsigned char* keep,
                        int E, int N) {
  int e = blockIdx.x * blockDim.x + threadIdx.x;
  if (e >= E) return;
  int s = ei[e], d = ei[E + e];
  unsigned char kp = 0;
  if (s != d) {
    long long idx = (long long)s * N + d;
    unsigned int bit = 1u << (idx & 31);
    unsigned int old = atomicOr(&bitmap[idx >> 5], bit);
    kp = (old & bit) ? 0 : 1;
  }
  keep[e] = kp;
}

// ---------------- conv1: one wave per edge, g1 params (16 channels) ----------------

__global__ void k_gconv1(const int* ei, const unsigned char* keep, const float* pos,
                         const float* w0, const float* b0, const float* w1,
                         const float* b1, const float* w2, const float* b2,
                         const float* wl, const float* bl, float* out1,
                         int E, int N, int total) {
  int slot = blockIdx.x * 8 + (threadIdx.x >> 5);
  if (slot >= total) return;                     // wave-uniform
  int s, d;
  if (!slot_edge(ei, keep, E, slot, s, d)) return;  // wave-uniform (per-slot flag)

  int lane = threadIdx.x & 31;
  bool hi = lane >= 16;
  int col = lane & 15;          // channel c (B/D column)
  int hb = hi ? 2 : 0;          // K sub-offset for this lane half
  int g8 = hi ? 8 : 0;          // D row offset for this lane half

  float f0, f1, f2;
  edge_feats(pos, s, d, f0, f1, f2);

  float cf = (float)col;
  v2f A1[4], A2[4], B1[4];
#pragma unroll
  for (int q = 0; q < 4; ++q) {
    int h0 = 4 * q + hb, h1 = h0 + 1;
    float base0 = f0 * w0[h0 * 4 + 0] + f1 * w0[h0 * 4 + 1] + f2 * w0[h0 * 4 + 2] + b0[h0];
    float base1 = f0 * w0[h1 * 4 + 0] + f1 * w0[h1 * 4 + 1] + f2 * w0[h1 * 4 + 2] + b0[h1];
    B1[q].x = sinf(OMEGA_F * (base0 + cf * w0[h0 * 4 + 3]));  // h1^T[h,c]
    B1[q].y = sinf(OMEGA_F * (base1 + cf * w0[h1 * 4 + 3]));
    A1[q].x = w1[col * 16 + h0]; A1[q].y = w1[col * 16 + h1]; // w1[g,h], g=col
    A2[q].x = w2[col * 16 + h0]; A2[q].y = w2[col * 16 + h1];
  }

  v8f acc = {};
#pragma unroll
  for (int q = 0; q < 4; ++q) acc = wmma4(A1[q], B1[q], acc);  // h2pre^T = w1 * h1^T

  float h2[8], S[8];
#pragma unroll
  for (int r = 0; r < 8; ++r) h2[r] = sinf(OMEGA_F * (acc[r] + b1[r + g8]));
#pragma unroll
  for (int r = 0; r < 8; ++r) S[r] = __shfl_xor(h2[r], 16, 32);  // lane-half swap

  v2f B2[4];   // repack D-layout h2^T into B-operand chunks
#pragma unroll
  for (int q = 0; q < 4; ++q) {
    int p = (q & 1) * 4 + hb;
    bool own = ((q >= 2) == hi);
    B2[q].x = own ? h2[p] : S[p];
    B2[q].y = own ? h2[p + 1] : S[p + 1];
  }

  v8f acc2 = {};
#pragma unroll
  for (int q = 0; q < 4; ++q) acc2 = wmma4(A2[q], B2[q], acc2); // h3pre^T = w2 * h2^T

  float ps0 = pos[s * 3 + 0], ps1 = pos[s * 3 + 1], ps2 = pos[s * 3 + 2];
  float part = 0.0f;
#pragma unroll
  for (int r = 0; r < 8; ++r) {
    int g = r + g8;
    float h3 = sinf(OMEGA_F * (acc2[r] + b2[g]));
    float v = wl[g] * ps0 + wl[16 + g] * ps1 + wl[32 + g] * ps2; // (wl^T pos_src)[g]
    part += h3 * v;
  }
  part += __shfl_xor(part, 16, 32);
  if (!hi) {
    float msg = part + bl[0] * ps0 + bl[1] * ps1 + bl[2] * ps2;
    atomicAdd(&out1[(long long)d * 16 + col], msg);
  }
}

// ---------------- selu(out1 + g1_bias) ----------------

__global__ void k_selu(const float* out1, const float* g1bias, float* hsel, int n) {
  int i = blockIdx.x * blockDim.x + threadIdx.x;
  if (i >= n) return;
  float v = out1[i] + g1bias[i & 15];
  hsel[i] = 1.0507009873554805f *
            (v > 0.0f ? v : 1.6732632423543772f * (expf(v) - 1.0f));
}

// ---------------- conv2: one wave per 16 edges, g2 params (1 channel) ----------------

__global__ void k_gconv2(const int* ei, const unsigned char* keep, const float* pos,
                         const float* hsel, const float* w0, const float* b0,
                         const float* w1, const float* b1, const float* w2,
                         const float* b2, const float* wl, const float* bl,
                         float* score, int E, int N, int total) {
  int wavebase = (blockIdx.x * 8 + (threadIdx.x >> 5)) * 16;
  if (wavebase >= total) return;                 // wave-uniform
  int lane = threadIdx.x & 31;
  bool hi = lane >= 16;
  int col = lane & 15;          // which edge of the 16 this lane serves
  int hb = hi ? 2 : 0;
  int g8 = hi ? 8 : 0;

  int slot = wavebase + col;
  int s = 0, d = 0;
  bool valid = (slot < total) && slot_edge(ei, keep, E, slot, s, d);
  if (!valid) { s = 0; d = 0; }                  // lanes stay active for WMMA

  float f0, f1, f2;
  edge_feats(pos, s, d, f0, f1, f2);

  v2f A1[4], A2[4], A3[4], B1[4], B3[4];
#pragma unroll
  for (int q = 0; q < 4; ++q) {
    int h0 = 4 * q + hb, h1 = h0 + 1;
    float base0 = f0 * w0[h0 * 4 + 0] + f1 * w0[h0 * 4 + 1] + f2 * w0[h0 * 4 + 2] + b0[h0];
    float base1 = f0 * w0[h1 * 4 + 0] + f1 * w0[h1 * 4 + 1] + f2 * w0[h1 * 4 + 2] + b0[h1];
    B1[q].x = sinf(OMEGA_F * base0);             // chan = 0 only
    B1[q].y = sinf(OMEGA_F * base1);
    A1[q].x = w1[col * 16 + h0]; A1[q].y = w1[col * 16 + h1];
    A2[q].x = w2[col * 16 + h0]; A2[q].y = w2[col * 16 + h1];
    A3[q].x = wl[h0 * 16 + col]; A3[q].y = wl[h1 * 16 + col]; // wl2^T[h,i], h=col
    B3[q].x = hsel[(long long)s * 16 + h0];                    // Hsel[i, edge]
    B3[q].y = hsel[(long long)s * 16 + h1];
  }

  v8f acc = {};
#pragma unroll
  for (int q = 0; q < 4; ++q) acc = wmma4(A1[q], B1[q], acc);

  float h2[8], S[8];
#pragma unroll
  for (int r = 0; r < 8; ++r) h2[r] = sinf(OMEGA_F * (acc[r] + b1[r + g8]));
#pragma unroll
  for (int r = 0; r < 8; ++r) S[r] = __shfl_xor(h2[r], 16, 32);

  v2f B2[4];
#pragma unroll
  for (int q = 0; q < 4; ++q) {
    int p = (q & 1) * 4 + hb;
    bool own = ((q >= 2) == hi);
    B2[q].x = own ? h2[p] : S[p];
    B2[q].y = own ? h2[p + 1] : S[p + 1];
  }

  v8f acc2 = {};
#pragma unroll
  for (int q = 0; q < 4; ++q) acc2 = wmma4(A2[q], B2[q], acc2);

  float h3[8];
#pragma unroll
  for (int r = 0; r < 8; ++r) h3[r] = sinf(OMEGA_F * (acc2[r] + b2[r + g8]));

  v8f accU = {};                                  // U = wl2^T * Hsel
#pragma unroll
  for (int q = 0; q < 4; ++q) accU = wmma4(A3[q], B3[q], accU);

  float part = 0.0f;
#pragma unroll
  for (int r = 0; r < 8; ++r) part += h3[r] * accU[r];
#pragma unroll
  for (int q = 0; q < 4; ++q) {                   // bl2 . hsel[src] (split by lane half)
    int h0 = 4 * q + hb;
    part += bl[h0] * B3[q].x + bl[h0 + 1] * B3[q].y;
  }
  part += __shfl_xor(part, 16, 32);
  if (!hi && valid) atomicAdd(&score[d], part);
}

// ---------------- sort keys: ascending stable argsort of score ----------------

__global__ void k_keys(const float* score, unsigned long long* keys, int N, int SZ) {
  int i = blockIdx.x * blockDim.x + threadIdx.x;
  if (i >= SZ) return;
  if (i < N) {
    unsigned int u = __float_as_uint(score[i]);
    u = (u & 0x80000000u) ? ~u : (u | 0x80000000u);
    keys[i] = ((unsigned long long)u << 32) | (unsigned int)i;
  } else {
    keys[i] = 0xFFFFFFFFFFFFFFFFull;
  }
}

__global__ void k_bitonic(unsigned long long* keys, int SZ) {
  for (int k = 2; k <= SZ; k <<= 1) {
    for (int j = k >> 1; j > 0; j >>= 1) {
      for (int i = threadIdx.x; i < SZ; i += blockDim.x) {
        int ixj = i ^ j;
        if (ixj > i) {
          unsigned long long a = keys[i], b = keys[ixj];
          bool up = ((i & k) == 0);
          if ((a > b) == up) { keys[i] = b; keys[ixj] = a; }
        }
      }
      __syncthreads();
    }
  }
}

// ---------------- CSR of augmented edges by src ----------------

__global__ void k_deg(const int* ei, const unsigned char* keep, int* csroff,
                      int E, int total) {
  int slot = blockIdx.x * blockDim.x + threadIdx.x;
  if (slot >= total) return;
  int s, d;
  if (slot_edge(ei, keep, E, slot, s, d)) atomicAdd(&csroff[s], 1);
}

__global__ void k_scan(int* csroff, int* cursor, int N) {
  if (threadIdx.x | blockIdx.x) return;
  int run = 0;
  for (int n = 0; n < N; ++n) {
    int c = csroff[n];
    csroff[n] = run;
    cursor[n] = run;
    run += c;
  }
  csroff[N] = run;
}

__global__ void k_scatter(const int* ei, const unsigned char* keep, int* cursor,
                          int* csrdst, int E, int total) {
  int slot = blockIdx.x * blockDim.x + threadIdx.x;
  if (slot >= total) return;
  int s, d;
  if (slot_edge(ei, keep, E, slot, s, d)) {
    int p = atomicAdd(&cursor[s], 1);
    csrdst[p] = d;
  }
}

// ---------------- greedy pooling loop (single block; reference semantics) ----------------

__global__ void k_pool(const unsigned long long* keys, int SZ, const int* csroff,
                       const int* csrdst, int* clustered, int* cluster,
                       const float* x, const float* pos, float* xpool,
                       float* pospool, int* members, int* scalars, int N) {
  __shared__ int sh_node, sh_k, sh_mcnt, sh_cursor;
  int tid = threadIdx.x;
  if (tid == 0) { sh_cursor = 0; sh_k = 0; }
  __syncthreads();
  for (;;) {
    if (tid == 0) {
      int c = sh_cursor, node = -1;
      while (c < SZ) {
        unsigned int idx = (unsigned int)(keys[c] & 0xFFFFFFFFull);
        if (idx < (unsigned int)N && clustered[idx] == 0) { node = (int)idx; break; }
        ++c;
      }
      sh_cursor = c;
      sh_node = node;
      sh_mcnt = 0;
    }
    __syncthreads();
    int node = sh_node;
    if (node < 0) break;
    int k = sh_k;
    int beg = csroff[node], end = csroff[node + 1];
    for (int j = beg + tid; j < end; j += blockDim.x) {
      int dn = csrdst[j];
      if (atomicExch(&clustered[dn], 1) == 0) {
        cluster[dn] = k;
        int p = atomicAdd(&sh_mcnt, 1);
        members[p] = dn;
      }
    }
    __syncthreads();
    int mc = sh_mcnt;   // >=1: node's own self-loop guarantees membership
    for (int f = tid; f < 64; f += blockDim.x) {
      float mx = -3.402823466e38f;
      for (int i = 0; i < mc; ++i)
        mx = fmaxf(mx, x[(long long)members[i] * 64 + f]);
      xpool[(long long)k * 64 + f] = mx;
    }
    if (tid == 0) {
      pospool[k * 3 + 0] = pos[node * 3 + 0];
      pospool[k * 3 + 1] = pos[node * 3 + 1];
      pospool[k * 3 + 2] = pos[node * 3 + 2];
      sh_k = k + 1;
    }
    __syncthreads();
  }
  if (tid == 0) scalars[0] = sh_k;
}

// ---------------- edge_pool: unique (cluster[s],cluster[d]) pairs, cs!=cd ----------------

__global__ void k_edgemark(const int* ei, const int* cluster, const int* scalars,
                           unsigned int* bitmap, int E) {
  int e = blockIdx.x * blockDim.x + threadIdx.x;
  if (e >= E) return;
  int cs = cluster[ei[e]];
  int cd = cluster[ei[E + e]];
  if (cs != cd) {
    long long K = scalars[0];
    long long idx = (long long)cs * K + cd;
    atomicOr(&bitmap[idx >> 5], 1u << (idx & 31));
  }
}

__global__ void k_finalize(const unsigned int* bitmap, int* scalars,
                           const float* pospool, const float* score, float* out,
                           int N) {
  if (threadIdx.x | blockIdx.x) return;
  long long K = scalars[0];
  long long total = K * K;
  long long m = 0;
  for (long long i = 0; i < total; ++i)
    if ((bitmap[i >> 5] >> (i & 31)) & 1u) ++m;
  long long base = K * 64;
  long long j = 0;
  for (long long i = 0; i < total; ++i)
    if ((bitmap[i >> 5] >> (i & 31)) & 1u) {
      out[base + j] = (float)(i / K);        // row 0: cluster[src]
      out[base + m + j] = (float)(i % K);    // row 1: cluster[dst]
      ++j;
    }
  float* pp = out + base + 2 * m;
  for (long long i = 0; i < K * 3; ++i) pp[i] = pospool[i];
  float* so = pp + K * 3;
  for (int n = 0; n < N; ++n) so[n] = score[n];
  scalars[1] = (int)m;
}

// ---------------- host ----------------

extern "C" void kernel_launch(void* const* d_in, const int* in_sizes, int n_in,
                              void* d_out, int out_size, void* d_ws, size_t ws_size,
                              hipStream_t stream) {
  const float* x      = (const float*)d_in[0];
  const float* pos    = (const float*)d_in[1];
  const int*   ei     = (const int*)d_in[2];
  const float* g1w0   = (const float*)d_in[3];
  const float* g1b0   = (const float*)d_in[4];
  const float* g1w1   = (const float*)d_in[5];
  const float* g1b1   = (const float*)d_in[6];
  const float* g1w2   = (const float*)d_in[7];
  const float* g1b2   = (const float*)d_in[8];
  const float* g1wl   = (const float*)d_in[9];
  const float* g1bl   = (const float*)d_in[10];
  const float* g1bias = (const float*)d_in[11];
  const float* g2w0   = (const float*)d_in[12];
  const float* g2b0   = (const float*)d_in[13];
  const float* g2w1   = (const float*)d_in[14];
  const float* g2b1   = (const float*)d_in[15];
  const float* g2w2   = (const float*)d_in[16];
  const float* g2b2   = (const float*)d_in[17];
  const float* g2wl   = (const float*)d_in[18];
  const float* g2bl   = (const float*)d_in[19];
  const float* g2bias = (const float*)d_in[20];

  int N = in_sizes[1] / 3;
  int E = in_sizes[2] / 2;
  int total = E + N;
  int SZ = 1;
  while (SZ < N) SZ <<= 1;

  char* wsb = (char*)d_ws;
  size_t off = 0;
  auto carve = [&](size_t bytes) -> void* {
    void* p = wsb + off;
    off += (bytes + 63) & ~(size_t)63;
    return p;
  };
  float* out1 = (float*)carve((size_t)N * 16 * 4);
  float* hsel = (float*)carve((size_t)N * 16 * 4);
  float* score = (float*)carve((size_t)N * 4);
  unsigned char* keep = (unsigned char*)carve((size_t)E);
  unsigned long long* keys = (unsigned long long*)carve((size_t)SZ * 8);
  int* csroff = (int*)carve((size_t)(N + 1) * 4);
  int* cursor = (int*)carve((size_t)N * 4);
  int* csrdst = (int*)carve((size_t)total * 4);
  int* clustered = (int*)carve((size_t)N * 4);
  int* cluster = (int*)carve((size_t)N * 4);
  int* members = (int*)carve((size_t)N * 4);
  float* pospool = (float*)carve((size_t)N * 3 * 4);
  int* scalars = (int*)carve(64);
  size_t bmwords = ((size_t)N * (size_t)N + 31) / 32;
  unsigned int* bitmap = (unsigned int*)carve(bmwords * 4);
  (void)ws_size; (void)n_in; (void)out_size;

  hipMemsetAsync(bitmap, 0, bmwords * 4, stream);
  hipMemsetAsync(out1, 0, (size_t)N * 16 * 4, stream);
  hipMemsetAsync(csroff, 0, (size_t)(N + 1) * 4, stream);
  hipMemsetAsync(clustered, 0, (size_t)N * 4, stream);
  hipMemsetAsync(cluster, 0, (size_t)N * 4, stream);
  hipMemsetAsync(scalars, 0, 64, stream);

  const int TB = 256;
  k_score_init<<<(N + TB - 1) / TB, TB, 0, stream>>>(score, g2bias, N);
  k_dedup<<<(E + TB - 1) / TB, TB, 0, stream>>>(ei, bitmap, keep, E, N);
  k_gconv1<<<(total + 7) / 8, TB, 0, stream>>>(ei, keep, pos, g1w0, g1b0, g1w1,
                                               g1b1, g1w2, g1b2, g1wl, g1bl,
                                               out1, E, N, total);
  k_selu<<<(N * 16 + TB - 1) / TB, TB, 0, stream>>>(out1, g1bias, hsel, N * 16);
  int waves2 = (total + 15) / 16;
  k_gconv2<<<(waves2 + 7) / 8, TB, 0, stream>>>(ei, keep, pos, hsel, g2w0, g2b0,
                                                g2w1, g2b1, g2w2, g2b2, g2wl,
                                                g2bl, score, E, N, total);
  k_keys<<<(SZ + TB - 1) / TB, TB, 0, stream>>>(score, keys, N, SZ);
  k_bitonic<<<1, 1024, 0, stream>>>(keys, SZ);
  k_deg<<<(total + TB - 1) / TB, TB, 0, stream>>>(ei, keep, csroff, E, total);
  k_scan<<<1, 1, 0, stream>>>(csroff, cursor, N);
  k_scatter<<<(total + TB - 1) / TB, TB, 0, stream>>>(ei, keep, cursor, csrdst, E, total);
  k_pool<<<1, 256, 0, stream>>>(keys, SZ, csroff, csrdst, clustered, cluster, x,
                                pos, (float*)d_out, pospool, members, scalars, N);
  hipMemsetAsync(bitmap, 0, bmwords * 4, stream);
  k_edgemark<<<(E + TB - 1) / TB, TB, 0, stream>>>(ei, cluster, scalars, bitmap, E);
  k_finalize<<<1, 1, 0, stream>>>(bitmap, scalars, pospool, score, (float*)d_out, N);
}